// InducGEN_76201309766388
// MI455X (gfx1250) — compile-verified
//
#include <hip/hip_runtime.h>
#include <hip/hip_bf16.h>

typedef __attribute__((ext_vector_type(2))) float v2f;
typedef __attribute__((ext_vector_type(8))) float v8f;

// ---------------------------------------------------------------------------
// Kernel 1: deterministic compaction of edges with dst == unseen_index.
// Single 1024-thread block (32 waves, wave32). Ballot + block prefix keeps
// the compacted list in ascending edge order -> deterministic fp sum order.
// ws[0] = match count, ws[4..] = matching edge ids.
// ---------------------------------------------------------------------------
__global__ void __launch_bounds__(1024)
indugen_filter_kernel(const int* __restrict__ edge_index, int E2,
                      const int* __restrict__ unseen_ptr,
                      int* __restrict__ ws, int cap) {
    const int unseen = unseen_ptr[0];
    const int* __restrict__ dst = edge_index + E2;   // row 1 of edge_index
    const int tid  = threadIdx.x;
    const int lane = tid & 31;
    const int wave = tid >> 5;

    __shared__ int wcnt[32];
    __shared__ int woff[32];
    __shared__ int chunk_base;

    int running = 0;
    for (int base = 0; base < E2; base += 1024) {
        const int e = base + tid;
        const bool m = (e < E2) && (dst[e] == unseen);
        const unsigned mask = (unsigned)__ballot(m);   // wave32: low 32 bits
        if (lane == 0) wcnt[wave] = __popc(mask);
        __syncthreads();
        if (tid == 0) {
            int s = 0;
            for (int w = 0; w < 32; ++w) { woff[w] = s; s += wcnt[w]; }
            chunk_base = running;
        }
        __syncthreads();
        const int pos = chunk_base + woff[wave] + __popc(mask & ((1u << lane) - 1u));
        if (m && pos < cap) ws[4 + pos] = e;
        running = chunk_base + woff[31] + wcnt[31];
        __syncthreads();   // protect wcnt/woff before next iteration overwrites
    }
    if (tid == 0) ws[0] = running;
}

// ---------------------------------------------------------------------------
// Kernel 2: for the M selected edges (tiles of 16), compute
//   out[o] = sum_e sum_b att[type(e),b] * (feat_e @ basis_b)[o]
// with V_WMMA_F32_16X16X4_F32. One block, 8 waves; wave w owns output
// columns [16w, 16w+16). All 4 bases and all K chunks chain into one
// accumulator; the dst scatter-add collapses to a row-sum of the D tile.
// ---------------------------------------------------------------------------
__global__ void __launch_bounds__(256)
indugen_msg_wmma_kernel(const float* __restrict__ entity,
                        const float* __restrict__ relation,
                        const float* __restrict__ basis,
                        const float* __restrict__ att,
                        const int*   __restrict__ node_id,
                        const int*   __restrict__ edge_index,
                        const int*   __restrict__ edge_type,
                        const int*   __restrict__ rel_index,
                        int E2,
                        const int*   __restrict__ ws, int cap,
                        float*       __restrict__ out) {
    __shared__ float sf[16][260];   // 16 edge rows x 256 feat, padded: 260%64=4
    __shared__ float sc[16][4];     // per-edge attention coefficients

    int M = ws[0];
    if (M > cap) M = cap;

    const int tid  = threadIdx.x;
    const int wave = tid >> 5;          // 0..7 -> output column tile
    const int lane = tid & 31;
    const int row  = lane & 15;         // matrix row (edge slot) / column-in-tile
    const int half = lane >> 4;         // 0: K=k0,k0+1   1: K=k0+2,k0+3

    float total = 0.0f;                 // column sum for col = wave*16 + row

    for (int t = 0; t * 16 < M; ++t) {
        __syncthreads();                // previous tile fully consumed
        // ---- stage attention coefficients -------------------------------
        if (tid < 64) {
            const int j = tid >> 2, b = tid & 3;
            const int g = t * 16 + j;
            float c = 0.0f;
            if (g < M) {
                const int eid = ws[4 + g];
                c = att[edge_type[eid] * 4 + b];
            }
            sc[j][b] = c;
        }
        // ---- stage concat(entity[node_id[src]], relation[rel_index]) ----
        for (int idx = tid; idx < 16 * 256; idx += 256) {
            const int j = idx >> 8, i = idx & 255;
            const int g = t * 16 + j;
            float f = 0.0f;
            if (g < M) {
                const int eid = ws[4 + g];
                if (i < 128) {
                    const int ent = node_id[edge_index[eid]];      // src row 0
                    f = entity[(long long)ent * 128 + i];
                } else {
                    f = relation[rel_index[eid] * 128 + (i - 128)];
                }
            }
            sf[j][i] = f;
        }
        __syncthreads();

        // ---- WMMA: acc += (coeff_b * Feat) @ basis_b, b = 0..3 ----------
        v8f acc = {};
        for (int b = 0; b < 4; ++b) {
            const float c = sc[row][b];
            const float* __restrict__ bp0 =
                basis + (b * 256) * 128 + (wave * 16 + row);
            for (int k0 = 0; k0 < 256; k0 += 4) {
                const int kk = k0 + half * 2;
                v2f A;
                A.x = c * sf[row][kk];
                A.y = c * sf[row][kk + 1];
                v2f B;
                const float* __restrict__ bp = bp0 + kk * 128;
                B.x = bp[0];
                B.y = bp[128];
                acc = __builtin_amdgcn_wmma_f32_16x16x4_f32(
                          false, A, false, B, (short)0, acc, false, false);
            }
        }
        // ---- row-sum of D tile (all rows share dst == unseen_index) -----
        float s = acc[0] + acc[1] + acc[2] + acc[3] +
                  acc[4] + acc[5] + acc[6] + acc[7];
        s += __shfl_xor(s, 16, 32);     // combine M=0..7 and M=8..15 halves
        total += s;
    }

    if (lane < 16) out[wave * 16 + row] = total;   // writes all 128 outputs
}

// ---------------------------------------------------------------------------
extern "C" void kernel_launch(void* const* d_in, const int* in_sizes, int n_in,
                              void* d_out, int out_size, void* d_ws, size_t ws_size,
                              hipStream_t stream) {
    const float* entity    = (const float*)d_in[0];
    const float* relation  = (const float*)d_in[1];
    const float* basis     = (const float*)d_in[2];
    const float* att       = (const float*)d_in[3];
    const int*   node_id   = (const int*)  d_in[4];
    const int*   edge_idx  = (const int*)  d_in[5];
    const int*   edge_type = (const int*)  d_in[6];
    const int*   rel_index = (const int*)  d_in[7];
    const int*   unseen    = (const int*)  d_in[8];

    const int E2  = in_sizes[5] / 2;
    int*      ws  = (int*)d_ws;
    int       cap = (int)(ws_size / sizeof(int)) - 4;
    if (cap < 16) cap = 16;   // ws is preallocated; defensive floor

    indugen_filter_kernel<<<1, 1024, 0, stream>>>(edge_idx, E2, unseen, ws, cap);
    indugen_msg_wmma_kernel<<<1, 256, 0, stream>>>(
        entity, relation, basis, att, node_id, edge_idx, edge_type, rel_index,
        E2, ws, cap, (float*)d_out);
}